// ChunkAggregator_85590108275021
// MI455X (gfx1250) — compile-verified
//
#include <hip/hip_runtime.h>

#define B_    8
#define L_    8192
#define V_    32000
#define D_    128
#define BLK_  64
#define NB_   128                      // n_blocks = L/BLOCK
#define SEQ_  (NB_ + NB_ + L_)         // 8448

#define OFF_NEWSEQ 0
#define OFF_CATIDS ((size_t)B_ * SEQ_ * D_)          // 8,650,752
#define OFF_HIST   (OFF_CATIDS + (size_t)B_ * NB_)   // 8,651,776
#define HIST_ELEMS ((size_t)B_ * NB_ * V_)           // 32,768,000

typedef __attribute__((ext_vector_type(2))) float v2f;
typedef __attribute__((ext_vector_type(8))) float v8f;

// ---------------------------------------------------------------------------
// Kernel 1: zero the 128MB hist region (float4 grid-stride, HBM-stream rate)
// ---------------------------------------------------------------------------
__global__ void zero_hist_kernel(float* __restrict__ hist) {
  const size_t n4 = HIST_ELEMS / 4;
  float4* p = (float4*)hist;
  const float4 z = {0.f, 0.f, 0.f, 0.f};
  for (size_t i = (size_t)blockIdx.x * blockDim.x + threadIdx.x; i < n4;
       i += (size_t)gridDim.x * blockDim.x)
    p[i] = z;
}

// ---------------------------------------------------------------------------
// Kernel 2: token embedding gather (1 wave per token, float4 per lane) +
//           scatter count into hist (lane 0, global_atomic_add_f32)
// ---------------------------------------------------------------------------
__global__ void token_kernel(const int* __restrict__ tokens,
                             const float* __restrict__ token_W,
                             float* __restrict__ out) {
  const int g    = blockIdx.x * blockDim.x + threadIdx.x;
  const int ti   = g >> 5;          // token index 0..B*L-1
  const int lane = g & 31;
  const int b    = ti >> 13;        // / 8192
  const int l    = ti & 8191;
  const int tok  = tokens[ti];

  const float4 v =
      *(const float4*)(token_W + (size_t)tok * D_ + (size_t)lane * 4);
  float* dst = out + OFF_NEWSEQ +
               ((size_t)b * SEQ_ + 2 * NB_ + l) * D_ + (size_t)lane * 4;
  *(float4*)dst = v;

  if (lane == 0) {
    float* hist = out + OFF_HIST;
    atomicAdd(hist + ((size_t)b * NB_ + (l >> 6)) * V_ + tok, 1.0f);
  }
}

// ---------------------------------------------------------------------------
// Kernel 3: per 64-token block: cat_ids, cat_emb gather, num_emb row-sum
//           via V_WMMA_F32_16X16X4_F32 with A = ones (D[m,n] = sum_k B[k,n]).
//           256 threads = 8 waves; wave w handles 16-column chunk w of D=128.
// ---------------------------------------------------------------------------
__global__ void block_kernel(const int* __restrict__ tokens,
                             const float* __restrict__ cat_W,
                             const float* __restrict__ num_W,
                             float* __restrict__ out) {
  __shared__ int toks[BLK_];
  const int bn = blockIdx.x;       // 0..B*NB_-1
  const int b  = bn >> 7;
  const int n  = bn & (NB_ - 1);
  const int t  = threadIdx.x;      // 0..255

  if (t < BLK_) toks[t] = tokens[(size_t)b * L_ + (size_t)n * BLK_ + t];
  __syncthreads();

  // cat_emb + cat_ids
  const int cat_id = toks[0];
  if (t < D_)
    out[OFF_NEWSEQ + ((size_t)b * SEQ_ + n) * D_ + t] =
        cat_W[(size_t)cat_id * D_ + t];
  if (t == 0)
    out[OFF_CATIDS + (size_t)b * NB_ + n] = (float)cat_id;

  // num_emb chunk via WMMA (EXEC all-ones here: straight-line, all threads)
  const int wave = t >> 5;         // column chunk 0..7
  const int lane = t & 31;
  const int col  = lane & 15;      // N = lane mod 16
  const int half = lane >> 4;      // lane half selects which K rows we feed
  const float* Wp = num_W + (size_t)wave * 16 + col;

  v2f a; a[0] = 1.0f; a[1] = 1.0f;   // A = ones(16x4)
  v8f c = {};
  for (int kk = 0; kk < BLK_; kk += 4) {
    const int k0 = kk + half * 2;
    v2f bm;
    bm[0] = Wp[(size_t)toks[k0] * D_];
    bm[1] = Wp[(size_t)toks[k0 + 1] * D_];
    c = __builtin_amdgcn_wmma_f32_16x16x4_f32(
        /*neg_a=*/false, a, /*neg_b=*/false, bm,
        /*c_mod=*/(short)0, c, /*reuse_a=*/false, /*reuse_b=*/false);
  }
  // D row M=0 lives in c[0] of lanes 0..15 (N = lane)
  if (half == 0)
    out[OFF_NEWSEQ + ((size_t)b * SEQ_ + NB_ + n) * D_ + wave * 16 + col] =
        c[0];
}

// ---------------------------------------------------------------------------
extern "C" void kernel_launch(void* const* d_in, const int* in_sizes, int n_in,
                              void* d_out, int out_size, void* d_ws,
                              size_t ws_size, hipStream_t stream) {
  const int*   tokens  = (const int*)d_in[0];
  const float* cat_W   = (const float*)d_in[1];
  const float* num_W   = (const float*)d_in[2];
  const float* token_W = (const float*)d_in[3];
  float* out = (float*)d_out;

  // 1) zero hist (must precede the atomic scatter; same-stream order suffices)
  zero_hist_kernel<<<2048, 256, 0, stream>>>(out + OFF_HIST);

  // 2) token gather + hist scatter: B*L waves = 65536, 8 tokens per 256-block
  token_kernel<<<(B_ * L_ * 32) / 256, 256, 0, stream>>>(tokens, token_W, out);

  // 3) per-block cat/num embeddings (WMMA path)
  block_kernel<<<B_ * NB_, 256, 0, stream>>>(tokens, cat_W, num_W, out);
}